// ConstituencyMFVI_5652176961690
// MI455X (gfx1250) — compile-verified
//
#include <hip/hip_runtime.h>
#include <hip/hip_bf16.h>
#include <stdint.h>

typedef float v2f __attribute__((ext_vector_type(2)));
typedef float v8f __attribute__((ext_vector_type(8)));
typedef __attribute__((address_space(3))) float       lds_f;
typedef __attribute__((address_space(3))) const float lds_cf;
typedef __attribute__((address_space(3))) const v2f   lds_cv2f;

static constexpr int L         = 128;
static constexpr int MP_STRIDE = 132;            // padded row stride (dwords) -> conflict-free ds_load_b64
static constexpr int MP_ELEMS  = L * MP_STRIDE;  // 16896 floats = 66 KB (CDNA5 WGP has 320 KB LDS)

__device__ __forceinline__ float sigmoidf(float x) {
    return 1.0f / (1.0f + __expf(-x));
}

__global__ __launch_bounds__(256) void ConstituencyMFVI_kernel(
    const float* __restrict__ s_span,
    const float* __restrict__ s_pair,
    const unsigned char* __restrict__ mask,
    float* __restrict__ out)
{
    __shared__ float Mp[MP_ELEMS];  // s_pair[b,i,:,:] tile, rows padded to 132 dwords
    __shared__ float sv[L];         // sigmoid(q) vector
    __shared__ float yv[L];         // matvec result

    const int tid = threadIdx.x;         // 256 threads = 8 wave32
    const int bi  = blockIdx.x;          // b*128 + i
    const int i   = bi & (L - 1);

    // ---- Stage the contiguous 64 KB tile s_pair[b,i,:,:] into padded LDS via async DMA.
    // 4096 x b128 transfers; 16 per lane; ASYNCcnt tracks completion.
    const float* gsrc = s_pair + (size_t)bi * (L * L);
    #pragma unroll
    for (int c = 0; c < 16; ++c) {
        const int chunk = c * 256 + tid;
        const int r     = chunk >> 5;          // row 0..127
        const int col   = (chunk & 31) << 2;   // col 0,4,...,124
        uint32_t goff = (uint32_t)((r * L + col) * 4);
        uint32_t loff = (uint32_t)(uintptr_t)(lds_f*)&Mp[r * MP_STRIDE + col];
        asm volatile("global_load_async_to_lds_b128 %0, %1, %2"
                     :: "v"(loff), "v"(goff), "s"(gsrc) : "memory");
    }

    // Row-private state (thread j = tid < 128 owns output row j for this (b,i)).
    float spanj = 0.f, mj = 0.f;
    int ls = 0, rs = 0;
    if (tid < L) {
        spanj = s_span[(size_t)bi * L + tid];
        mj    = mask[(size_t)bi * L + tid] ? 1.0f : 0.0f;
        ls = min(i, tid);
        rs = max(i, tid);
        sv[tid] = sigmoidf(spanj);   // q0 = span
    }

    asm volatile("s_wait_asynccnt 0x0" ::: "memory");
    __syncthreads();                 // tile + sv visible to all waves

    // Mask correction coefficients (M is loop-invariant; only sv changes).
    float Mjls = 0.f, Mjrs = 0.f;
    if (tid < L) {
        Mjls = Mp[tid * MP_STRIDE + ls];
        Mjrs = (rs != ls) ? Mp[tid * MP_STRIDE + rs] : 0.0f;  // i==j: zero one entry, not two
    }

    // WMMA f32 16x16x4 lane mapping: lanes 0-15 -> rows M=0..15, K=0,1 in vgpr0/1;
    // lanes 16-31 -> same rows, K=2,3. B replicated across columns => every D column = y chunk.
    const int lane = tid & 31;
    const int half = lane >> 4;                    // 0: K=0,1   1: K=2,3
    const int row  = (tid >> 5) * 16 + (lane & 15);
    lds_cf* arow = (lds_cf*)&Mp[row * MP_STRIDE + 2 * half];
    lds_cf* bcol = (lds_cf*)&sv[2 * half];

    for (int it = 0; it < 3; ++it) {
        v8f acc = {};
        #pragma unroll
        for (int k = 0; k < L; k += 4) {
            v2f a = *(lds_cv2f*)(arow + k);        // ds_load_b64: M[row, k+2h : k+2h+1]
            v2f b = *(lds_cv2f*)(bcol + k);        // ds_load_b64: sv[k+2h : k+2h+1] (broadcast)
            acc = __builtin_amdgcn_wmma_f32_16x16x4_f32(
                false, a, false, b, (short)0, acc, false, false);
        }
        // All D columns identical: lane0 holds y[w16+0..7] in acc[0..7], lane16 holds y[w16+8..15].
        if ((lane & 15) == 0) {
            const int ybase = (tid >> 5) * 16 + half * 8;
            #pragma unroll
            for (int v = 0; v < 8; ++v) yv[ybase + v] = acc[v];
        }
        __syncthreads();

        float m = 0.f;
        if (tid < L) {
            float corr = Mjls * sv[ls] + Mjrs * sv[rs];     // undo the two unmasked terms
            float q    = spanj + mj * (yv[tid] - corr);     // row gate by mask
            m = sigmoidf(q);
        }
        __syncthreads();                  // all reads of old sv complete
        if (tid < L) {
            if (it < 2) sv[tid] = m;
            else        out[(size_t)bi * L + tid] = m;      // marginals = sigmoid(q)
        }
        __syncthreads();                  // sv updated before next matvec
    }
}

extern "C" void kernel_launch(void* const* d_in, const int* in_sizes, int n_in,
                              void* d_out, int out_size, void* d_ws, size_t ws_size,
                              hipStream_t stream) {
    const float* s_span        = (const float*)d_in[0];
    const float* s_pair        = (const float*)d_in[1];
    const unsigned char* maskp = (const unsigned char*)d_in[2];
    float* out                 = (float*)d_out;

    const int B = in_sizes[0] / (L * L);     // 32
    dim3 grid(B * L);                        // one workgroup per (b,i): 4096 blocks
    dim3 block(256);                         // 8 wave32
    ConstituencyMFVI_kernel<<<grid, block, 0, stream>>>(s_span, s_pair, maskp, out);
}